// TTTLinearLayer_65687229825786
// MI455X (gfx1250) — compile-verified
//
#include <hip/hip_runtime.h>

// TTT-Linear (chunked delta-rule) for MI455X / gfx1250.
// f32 WMMA everywhere + Tensor Data Mover double-buffered LDS staging.
// D=512, B=8, T=2048. Inputs: x, W0, Wk, Wv(unused), Wq, ln_g, ln_b. Out f32 (B,T,D).

#define DD    512
#define BATCH   8
#define TSEQ 2048
#define LR    0.01f
#define EPS   1e-5f
#define LDSROW 516           // effective row stride (floats) incl. padding
#define CHUNKF (16 * LDSROW) // floats per staged 16x512 chunk buffer

typedef float v2f __attribute__((ext_vector_type(2)));
typedef float v8f __attribute__((ext_vector_type(8)));
typedef unsigned int u32x4 __attribute__((ext_vector_type(4)));
typedef int i32x4 __attribute__((ext_vector_type(4)));
typedef int i32x8 __attribute__((ext_vector_type(8)));

// TDM pad: pad_interval=7 (every 256 dwords) + pad_amount=1 (2 dwords)
// => row stride 516 floats; element (row,d) lives at row*516 + d + 2*(d>>8).
__device__ __forceinline__ int kqidx(int row, int dd) {
  return row * LDSROW + dd + (((dd) >> 8) << 1);
}

// D(16x16,f32) = A(16x4,f32) * B(4x16,f32) + C        (wave32)
// Lane l: m/n = l&15, kb = 2*(l>>4).
//   A: {A[m][kb], A[m][kb+1]}   B: {B[kb][n], B[kb+1][n]}   C/D: c[i]=C[i+8*(l>>4)][n]
__device__ __forceinline__ v8f wmma4(v2f a, v2f b, v8f c) {
  return __builtin_amdgcn_wmma_f32_16x16x4_f32(false, a, false, b, (short)0, c,
                                               false, false);
}

// Issue one TDM load of a 16x512 f32 tile (row stride 512 floats in memory)
// into LDS at byte offset lds_off, with the padding described above.
__device__ __forceinline__ void tdm_load_chunk(unsigned lds_off,
                                               const float* gaddr) {
  const unsigned long long ga = (unsigned long long)(size_t)gaddr;
  u32x4 g0;
  g0[0] = 1u;                                        // count=1, user descriptor
  g0[1] = lds_off;                                   // lds_addr (bytes)
  g0[2] = (unsigned)(ga & 0xFFFFFFFFull);            // global_addr[31:0]
  g0[3] = (unsigned)((ga >> 32) & 0x1FFFFFFull)      // global_addr[56:32]
          | 0x80000000u;                             // type=2 ("image")
  i32x8 g1;
  g1[0] = (int)((2u << 16)      // data_size = 4 bytes
                | (1u << 20)    // pad_enable
                | (7u << 22)    // pad_interval: every 256 dwords
                | (1u << 25));  // pad_amount: 2 dwords
  g1[1] = (int)(512u << 16);    // tensor_dim0 = 512
  g1[2] = (int)(16u << 16);     // tensor_dim1 = 16
  g1[3] = (int)(512u << 16);    // tile_dim0 = 512
  g1[4] = 16;                   // tile_dim1 = 16 (tile_dim2 = 0 -> 2D)
  g1[5] = 512;                  // tensor_dim0_stride = 512 elements
  g1[6] = 0;
  g1[7] = 0;
  const i32x4 z4 = {0, 0, 0, 0};
#if defined(__clang_major__) && __clang_major__ >= 23
  const i32x8 z8 = {0, 0, 0, 0, 0, 0, 0, 0};
  __builtin_amdgcn_tensor_load_to_lds(g0, g1, z4, z4, z8, 0);
#else
  __builtin_amdgcn_tensor_load_to_lds(g0, g1, z4, z4, 0);
#endif
}

// ---------------------------------------------------------------------------
// Kernel 1: K = x @ Wk^T, Q = x @ Wq^T.  One wave per 16x16 output tile of
// both K and Q (shared A operand).  grid = (D/16, B*T/16), block = 32.
// ---------------------------------------------------------------------------
__global__ void ttt_proj_kernel(const float* __restrict__ x,
                                const float* __restrict__ Wk,
                                const float* __restrict__ Wq,
                                float* __restrict__ Kout,
                                float* __restrict__ Qout) {
  const int e0   = blockIdx.x * 16;
  const int row0 = blockIdx.y * 16;
  const int lane = threadIdx.x;
  const int m  = lane & 15;
  const int h  = lane >> 4;
  const int kb = h * 2;

  const float* xrow = x  + (size_t)(row0 + m) * DD;
  const float* krow = Wk + (size_t)(e0 + m)  * DD;
  const float* qrow = Wq + (size_t)(e0 + m)  * DD;

  v8f cK = {}; v8f cQ = {};
  for (int d = 0; d < DD; d += 4) {
    v2f a  = *(const v2f*)(xrow + d + kb);
    v2f bk = *(const v2f*)(krow + d + kb);
    v2f bq = *(const v2f*)(qrow + d + kb);
    cK = wmma4(a, bk, cK);
    cQ = wmma4(a, bq, cQ);
  }
  for (int i = 0; i < 8; ++i) {
    const size_t rr = (size_t)(row0 + i + 8 * h) * DD + e0 + m;
    Kout[rr] = cK[i];
    Qout[rr] = cQ[i];
  }
}

// ---------------------------------------------------------------------------
// Kernel 2: chunked TTT scan.  grid = (D/16 row-blocks, B), block = 32 (1 wave).
// dW = (W - W0) 16-row slice lives in LDS across all T/16 chunks.  K/Q chunks
// are TDM double-buffered.  Per chunk (C=16):
//   S = Kc Kc^T,  M = tril(Qc Kc^T),  A0 = Kc dW^T        (WMMA)
//   E[t] = A0[t] - LR * sum_{s<t} S[t][s] E[s]            (16-step solve)
//   O  = Qc (W0+dW)^T - LR * M E                          (WMMA, chunk-start dW)
//   dW = dW - LR * E^T Kc                                 (WMMA)
// ---------------------------------------------------------------------------
__global__ void ttt_scan_kernel(const float* __restrict__ Kin,
                                const float* __restrict__ Qin,
                                const float* __restrict__ W0,
                                float* __restrict__ Out) {
  extern __shared__ float smem[];
  float* dW  = smem;                       // 16 * LDSROW (plain row stride 516)
  float* KQ  = dW + 16 * LDSROW;           // 4 chunk buffers: K0 Q0 K1 Q1
  float* Ss  = KQ + 4 * CHUNKF;            // 16*16
  float* Ms  = Ss + 256;                   // 16*16 (masked)
  float* A0s = Ms + 256;                   // 16*16
  float* Es  = A0s + 256;                  // 16*16

  const int rblk = blockIdx.x;
  const int b    = blockIdx.y;
  const int lane = threadIdx.x;
  const int n  = lane & 15;
  const int h  = lane >> 4;
  const int kb = h * 2;
  const int r0 = rblk * 16;

  for (int i = lane; i < 16 * LDSROW; i += 32) dW[i] = 0.0f;
  __syncthreads();

  const float* Kb = Kin + (size_t)b * TSEQ * DD;
  const float* Qb = Qin + (size_t)b * TSEQ * DD;
  float*       Ob = Out + (size_t)b * TSEQ * DD;
  const float* w0row = W0 + (size_t)(r0 + n) * DD;

  const unsigned kqBase = (unsigned)(size_t)KQ;   // LDS byte offset of buffers

  // Prefetch chunk 0 into buffer set 0 via the Tensor Data Mover.
  tdm_load_chunk(kqBase + 0u * CHUNKF * 4u, Kb);
  tdm_load_chunk(kqBase + 1u * CHUNKF * 4u, Qb);

  const int NCH = TSEQ / 16;
  for (int c = 0; c < NCH; ++c) {
    const int t0  = c * 16;
    const int cur = c & 1;
    float* Ks = KQ + (2 * cur + 0) * CHUNKF;
    float* Qs = KQ + (2 * cur + 1) * CHUNKF;

    if (c + 1 < NCH) {
      // Target buffers were last *read* in iteration c-1: drain DS first.
      asm volatile("s_wait_dscnt 0x0" ::: "memory");
      const int nxt = (c + 1) & 1;
      tdm_load_chunk(kqBase + (unsigned)(2 * nxt + 0) * CHUNKF * 4u,
                     Kb + (size_t)(t0 + 16) * DD);
      tdm_load_chunk(kqBase + (unsigned)(2 * nxt + 1) * CHUNKF * 4u,
                     Qb + (size_t)(t0 + 16) * DD);
      // 4 outstanding, in-order: <=2 left means current chunk has landed.
      asm volatile("s_wait_tensorcnt 0x2" ::: "memory");
    } else {
      asm volatile("s_wait_tensorcnt 0x0" ::: "memory");
    }

    // ---- S = K K^T, M = Q K^T, A0 = K dW^T (accumulate over D) ----
    v8f cS = {}; v8f cM = {}; v8f cA = {};
    for (int d = 0; d < DD; d += 4) {
      const int ki = kqidx(n, d + kb);
      v2f ak = { Ks[ki],     Ks[ki + 1] };
      v2f aq = { Qs[ki],     Qs[ki + 1] };
      v2f bw = { dW[n * LDSROW + d + kb], dW[n * LDSROW + d + kb + 1] };
      cS = wmma4(ak, ak, cS);   // A and B register images coincide for K K^T
      cM = wmma4(aq, ak, cM);
      cA = wmma4(ak, bw, cA);
    }
    for (int i = 0; i < 8; ++i) {
      const int row = i + 8 * h;                      // time index in chunk
      Ss [row * 16 + n] = cS[i];
      A0s[row * 16 + n] = cA[i];
      Ms [row * 16 + n] = (n <= row) ? cM[i] : 0.0f;  // inclusive tril mask
    }
    __syncthreads();

    // ---- sequential triangular solve for E (16 lanes, one column each) ----
    if (lane < 16) {
      for (int t = 0; t < 16; ++t) {
        float e = A0s[t * 16 + lane];
        for (int s = 0; s < t; ++s) e -= LR * Ss[t * 16 + s] * Es[s * 16 + lane];
        Es[t * 16 + lane] = e;
      }
    }
    __syncthreads();

    // ---- O = Q (W0 + dW)^T - LR * tril(M) E  (uses chunk-start dW) ----
    v8f cO = {};
    for (int d = 0; d < DD; d += 4) {
      const int qi = kqidx(n, d + kb);
      v2f aq = { Qs[qi], Qs[qi + 1] };
      v2f bw = { w0row[d + kb]     + dW[n * LDSROW + d + kb],
                 w0row[d + kb + 1] + dW[n * LDSROW + d + kb + 1] };
      cO = wmma4(aq, bw, cO);
    }
    for (int s0 = 0; s0 < 16; s0 += 4) {
      v2f am = { Ms[n * 16 + s0 + kb], Ms[n * 16 + s0 + kb + 1] };
      v2f be = { -LR * Es[(s0 + kb) * 16 + n], -LR * Es[(s0 + kb + 1) * 16 + n] };
      cO = wmma4(am, be, cO);
    }
    for (int i = 0; i < 8; ++i)
      Ob[(size_t)(t0 + i + 8 * h) * DD + r0 + n] = cO[i];

    // ---- dW -= LR * E^T K  (tile over D) ----
    for (int d0 = 0; d0 < DD; d0 += 16) {
      v8f cW;
      for (int i = 0; i < 8; ++i) cW[i] = dW[(i + 8 * h) * LDSROW + d0 + n];
      for (int s0 = 0; s0 < 16; s0 += 4) {
        v2f ae = { -LR * Es[(s0 + kb) * 16 + n], -LR * Es[(s0 + kb + 1) * 16 + n] };
        v2f bk = { Ks[kqidx(s0 + kb, d0 + n)], Ks[kqidx(s0 + kb + 1, d0 + n)] };
        cW = wmma4(ae, bk, cW);
      }
      for (int i = 0; i < 8; ++i) dW[(i + 8 * h) * LDSROW + d0 + n] = cW[i];
    }
    __syncthreads();
  }
}

// ---------------------------------------------------------------------------
// Kernel 3: in-place LayerNorm over last dim.  One wave per (b,t) row.
// ---------------------------------------------------------------------------
__global__ void ttt_ln_kernel(float* __restrict__ out,
                              const float* __restrict__ g,
                              const float* __restrict__ bta) {
  const int lane = threadIdx.x;
  float* p = out + (size_t)blockIdx.x * DD;
  float v[16];
  float s = 0.0f, s2 = 0.0f;
  for (int j = 0; j < 16; ++j) {
    v[j] = p[lane + 32 * j];
    s  += v[j];
    s2 += v[j] * v[j];
  }
  for (int m = 16; m; m >>= 1) {
    s  += __shfl_xor(s,  m, 32);
    s2 += __shfl_xor(s2, m, 32);
  }
  const float mean = s * (1.0f / DD);
  const float var  = s2 * (1.0f / DD) - mean * mean;
  const float inv  = rsqrtf(var + EPS);
  for (int j = 0; j < 16; ++j) {
    const int c = lane + 32 * j;
    p[c] = (v[j] - mean) * inv * g[c] + bta[c];
  }
}

// ---------------------------------------------------------------------------
extern "C" void kernel_launch(void* const* d_in, const int* in_sizes, int n_in,
                              void* d_out, int out_size, void* d_ws, size_t ws_size,
                              hipStream_t stream) {
  (void)in_sizes; (void)n_in; (void)out_size; (void)ws_size;
  const float* x   = (const float*)d_in[0];
  const float* W0  = (const float*)d_in[1];
  const float* Wk  = (const float*)d_in[2];
  // d_in[3] = Wv: computed by the torch module but unused in the update.
  const float* Wq  = (const float*)d_in[4];
  const float* gln = (const float*)d_in[5];
  const float* bln = (const float*)d_in[6];
  float* out = (float*)d_out;

  float* Kw = (float*)d_ws;                        // B*T*D f32
  float* Qw = Kw + (size_t)BATCH * TSEQ * DD;      // B*T*D f32

  ttt_proj_kernel<<<dim3(DD / 16, BATCH * TSEQ / 16), 32, 0, stream>>>(
      x, Wk, Wq, Kw, Qw);

  const size_t smem =
      (size_t)(16 * LDSROW + 4 * CHUNKF + 4 * 256) * sizeof(float);
  ttt_scan_kernel<<<dim3(DD / 16, BATCH), 32, smem, stream>>>(Kw, Qw, W0, out);

  ttt_ln_kernel<<<BATCH * TSEQ, 32, 0, stream>>>(out, gln, bln);
}